// Decoder_84825604096733
// MI455X (gfx1250) — compile-verified
//
#include <hip/hip_runtime.h>
#include <hip/hip_bf16.h>
#include <math.h>

// Problem constants
#define BB   32
#define TT   64
#define SS   64
#define UU   512
#define FOURU 2048
#define HSTRIDE 516   // padded LDS row stride (floats) -> conflict-free column reads

typedef __attribute__((ext_vector_type(2))) float v2f;
typedef __attribute__((ext_vector_type(8))) float v8f;

// V_WMMA_F32_16X16X4_F32 : D(16x16 f32) = A(16x4 f32) * B(4x16 f32) + C
// 8-arg form: (neg_a, A, neg_b, B, c_mod, C, reuse_a, reuse_b)
__device__ __forceinline__ v8f wmma_f32_16x16x4(v2f a, v2f b, v8f c) {
  return __builtin_amdgcn_wmma_f32_16x16x4_f32(false, a, false, b, (short)0, c,
                                               false, false);
}

__device__ __forceinline__ float sigf(float x) { return 1.0f / (1.0f + expf(-x)); }

// Recover the raw LDS byte offset of a __shared__ pointer: generic->local
// addrspacecast strips the aperture, leaving the group-segment offset.
__device__ __forceinline__ unsigned lds_off(const void* p) {
  return (unsigned)(size_t)(__attribute__((address_space(3))) const char*)p;
}

// CDNA5 async global->LDS copy of 16 bytes per lane (ASYNCcnt-tracked).
// dst = LDS byte offset (per lane), src = 64-bit global address (per lane).
__device__ __forceinline__ void async_copy_b128(unsigned dst, const void* src) {
  asm volatile("global_load_async_to_lds_b128 %0, %1, off"
               :: "v"(dst), "v"(src) : "memory");
}
__device__ __forceinline__ void async_wait0() {
  asm volatile("s_wait_asynccnt 0x0" ::: "memory");
}

// ---------------------------------------------------------------------------
// Kernel 1: xz[b,t,:] = Wx[label[b,t],:] + bias   (one-hot matmul == gather)
// ---------------------------------------------------------------------------
__global__ __launch_bounds__(256) void xz_gather_kernel(
    const int* __restrict__ labels,   // [B*T]
    const float* __restrict__ Wx,     // [VOCAB, 4U]
    const float* __restrict__ bias,   // [4U]
    float* __restrict__ xz)           // [B*T, 4U]
{
  const int bt = blockIdx.x;
  const int lbl = labels[bt];
  const float* src = Wx + (size_t)lbl * FOURU;
  float* dst = xz + (size_t)bt * FOURU;
  for (int i = threadIdx.x; i < FOURU; i += 256)
    dst[i] = src[i] + bias[i];
}

// ---------------------------------------------------------------------------
// Kernel 2: LSTM over T steps.  grid.x = 2 (16 batch rows each = 1 M-tile),
// 512 threads = 16 waves; wave w owns u-columns [32w, 32w+32) for all 4 gates
// so c-state and gate fusion stay in registers.  h lives in LDS.
// ---------------------------------------------------------------------------
__global__ __launch_bounds__(512) void lstm_kernel(
    const float* __restrict__ xz,   // [B,T,4U]
    const float* __restrict__ Wh,   // [U,4U]
    const float* __restrict__ h0,   // [B,U]
    const float* __restrict__ c0,   // [B,U]
    float* __restrict__ dec)        // [B,T,U]
{
  __shared__ float hbuf[16 * HSTRIDE];       // 33 KB, padded rows

  const int bm   = blockIdx.x;               // batch-row group (0..1)
  const int tid  = threadIdx.x;
  const int wave = tid >> 5;
  const int lane = tid & 31;
  const int lo   = lane & 15;                // N (or M for A-frag) index
  const int hi   = lane >> 4;                // lane-half select
  const int ubase = wave * 32;               // this wave's u-column base

  // init h in LDS via async global->LDS copies (4-float chunks, padded rows)
  for (int c4 = tid; c4 < (16 * UU) / 4; c4 += 512) {
    int r = c4 >> 7;                 // 128 chunks per 512-float row
    int c = (c4 & 127) * 4;
    async_copy_b128(lds_off(&hbuf[r * HSTRIDE + c]),
                    &h0[(size_t)(bm * 16 + r) * UU + c]);
  }

  // c state in registers, laid out exactly like the WMMA C/D fragment
  v8f creg[2];
  for (int uj = 0; uj < 2; ++uj) {
    const int u = ubase + uj * 16 + lo;
    for (int r = 0; r < 8; ++r) {
      int m = r + hi * 8;
      creg[uj][r] = c0[(size_t)(bm * 16 + m) * UU + u];
    }
  }
  async_wait0();
  __syncthreads();

  for (int t = 0; t < TT; ++t) {
    // accumulators: [u-tile][gate], seeded with xz (input projection + bias)
    v8f acc[2][4];
    for (int uj = 0; uj < 2; ++uj)
      for (int g = 0; g < 4; ++g)
        for (int r = 0; r < 8; ++r) {
          int m = r + hi * 8;
          acc[uj][g][r] =
              xz[((size_t)(bm * 16 + m) * TT + t) * FOURU + g * UU + ubase + uj * 16 + lo];
        }

    // z += h @ Wh, K = 512 in steps of 4; A-frag reused for all 8 tiles
    const int kh = hi * 2;
    for (int k = 0; k < UU; k += 4) {
      v2f a;
      a.x = hbuf[lo * HSTRIDE + k + kh];
      a.y = hbuf[lo * HSTRIDE + k + kh + 1];
      if (k + 8 < UU)
        __builtin_prefetch(&Wh[(size_t)(k + 8) * FOURU + ubase + lo], 0, 1);
#pragma unroll
      for (int uj = 0; uj < 2; ++uj)
#pragma unroll
        for (int g = 0; g < 4; ++g) {
          const int col = g * UU + ubase + uj * 16 + lo;
          v2f bfrag;
          bfrag.x = Wh[(size_t)(k + kh) * FOURU + col];
          bfrag.y = Wh[(size_t)(k + kh + 1) * FOURU + col];
          acc[uj][g] = wmma_f32_16x16x4(a, bfrag, acc[uj][g]);
        }
    }
    __syncthreads();   // all waves done READING h before anyone overwrites it

    // gate math (keras order i,f,g,o), update c (regs) and h (LDS), emit dec
    for (int uj = 0; uj < 2; ++uj) {
      const int u = ubase + uj * 16 + lo;
#pragma unroll
      for (int r = 0; r < 8; ++r) {
        int m = r + hi * 8;
        float zi = acc[uj][0][r], zf = acc[uj][1][r];
        float zg = acc[uj][2][r], zo = acc[uj][3][r];
        float cc = sigf(zf) * creg[uj][r] + sigf(zi) * tanhf(zg);
        float hh = sigf(zo) * tanhf(cc);
        creg[uj][r] = cc;
        hbuf[m * HSTRIDE + u] = hh;
        dec[((size_t)(bm * 16 + m) * TT + t) * UU + u] = hh;
      }
    }
    __syncthreads();   // h fully written before next step reads it
  }
}

// ---------------------------------------------------------------------------
// Kernel 3: out[M,512] = A[M,512] @ W[512,512] + bias   (used for d1 and e2)
// Block: 256 threads = 8 waves, one 16x16 N-tile per wave; A-tile staged in
// LDS with CDNA5 async global->LDS copies.
// ---------------------------------------------------------------------------
__global__ __launch_bounds__(256) void gemm_bias_kernel(
    const float* __restrict__ A,     // [M,512]
    const float* __restrict__ W,     // [512,512]
    const float* __restrict__ bias,  // [512]
    float* __restrict__ out)         // [M,512]
{
  __shared__ float atile[16 * HSTRIDE];   // 33 KB

  const int tid = threadIdx.x, wave = tid >> 5, lane = tid & 31;
  const int lo = lane & 15, hi = lane >> 4;
  const int m0 = blockIdx.x * 16;
  const int n0 = (blockIdx.y * 8 + wave) * 16;

  // stage A tile: 2048 float4 chunks, 8 async b128 copies per thread
  for (int c4 = tid; c4 < (16 * UU) / 4; c4 += 256) {
    int r = c4 >> 7;
    int c = (c4 & 127) * 4;
    async_copy_b128(lds_off(&atile[r * HSTRIDE + c]),
                    &A[(size_t)(m0 + r) * UU + c]);
  }
  async_wait0();
  __syncthreads();

  v8f acc;
  {
    float bz = bias[n0 + lo];        // bias broadcast along M
    for (int r = 0; r < 8; ++r) acc[r] = bz;
  }

  const int kh = hi * 2;
  for (int k = 0; k < UU; k += 4) {
    v2f a, b;
    a.x = atile[lo * HSTRIDE + k + kh];
    a.y = atile[lo * HSTRIDE + k + kh + 1];
    b.x = W[(size_t)(k + kh) * UU + n0 + lo];
    b.y = W[(size_t)(k + kh + 1) * UU + n0 + lo];
    if (k + 8 < UU)
      __builtin_prefetch(&W[(size_t)(k + 8) * UU + n0 + lo], 0, 1);
    acc = wmma_f32_16x16x4(a, b, acc);
  }

#pragma unroll
  for (int r = 0; r < 8; ++r) {
    int m = r + hi * 8;
    out[(size_t)(m0 + m) * UU + n0 + lo] = acc[r];
  }
}

// ---------------------------------------------------------------------------
// Kernel 4: fused Bahdanau score + softmax, one block per (b,t).
// logits[s] = sum_u tanh(d1[b,t,u] + e2[b,s,u]) * V[u] + bv ; softmax over s;
// store transposed to out[B,S,T].
// ---------------------------------------------------------------------------
__global__ __launch_bounds__(256) void attn_softmax_kernel(
    const float* __restrict__ d1,   // [B*T, 512]
    const float* __restrict__ e2,   // [B*S, 512]
    const float* __restrict__ V,    // [512]
    const float* __restrict__ bv,   // [1]
    float* __restrict__ out)        // [B,S,T]
{
  __shared__ float drow[UU];
  __shared__ float vrow[UU];
  __shared__ float logits[SS];

  const int bt = blockIdx.x;
  const int b = bt >> 6, t = bt & 63;
  const int tid = threadIdx.x, wave = tid >> 5, lane = tid & 31;

  // stage d1 row (128 chunks) + V (128 chunks): one async b128 per thread
  {
    unsigned dst;
    const float* src;
    if (tid < 128) {
      dst = lds_off(&drow[tid * 4]);
      src = &d1[(size_t)bt * UU + tid * 4];
    } else {
      int i = (tid - 128) * 4;
      dst = lds_off(&vrow[i]);
      src = &V[i];
    }
    async_copy_b128(dst, src);
    async_wait0();
  }
  __syncthreads();

  const float bvv = bv[0];
  for (int s = wave; s < SS; s += 8) {
    const float* e2s = e2 + ((size_t)b * SS + s) * UU;
    float p = 0.0f;
    for (int u = lane; u < UU; u += 32)
      p += tanhf(drow[u] + e2s[u]) * vrow[u];
    for (int off = 16; off > 0; off >>= 1)
      p += __shfl_xor(p, off, 32);
    if (lane == 0) logits[s] = p + bvv;
  }
  __syncthreads();

  if (tid < 32) {
    float l0 = logits[tid], l1 = logits[tid + 32];
    float m = fmaxf(l0, l1);
    for (int off = 16; off > 0; off >>= 1)
      m = fmaxf(m, __shfl_xor(m, off, 32));
    float e0 = expf(l0 - m), e1 = expf(l1 - m);
    float ssum = e0 + e1;
    for (int off = 16; off > 0; off >>= 1)
      ssum += __shfl_xor(ssum, off, 32);
    float inv = 1.0f / ssum;
    out[((size_t)b * SS + tid) * TT + t]      = e0 * inv;   // [b, s, t]
    out[((size_t)b * SS + tid + 32) * TT + t] = e1 * inv;
  }
}

// ---------------------------------------------------------------------------
extern "C" void kernel_launch(void* const* d_in, const int* in_sizes, int n_in,
                              void* d_out, int out_size, void* d_ws, size_t ws_size,
                              hipStream_t stream) {
  const int*   labels = (const int*)  d_in[0];   // [B,T]
  const float* enc    = (const float*)d_in[1];   // [B,S,U]
  const float* h0     = (const float*)d_in[2];   // [B,U]
  const float* c0     = (const float*)d_in[3];   // [B,U]
  const float* Wx     = (const float*)d_in[4];   // [VOCAB,4U]
  const float* Wh     = (const float*)d_in[5];   // [U,4U]
  const float* bvec   = (const float*)d_in[6];   // [4U]
  const float* W1     = (const float*)d_in[7];   // [U,U]
  const float* b1     = (const float*)d_in[8];   // [U]
  const float* W2     = (const float*)d_in[9];   // [U,U]
  const float* b2     = (const float*)d_in[10];  // [U]
  const float* V      = (const float*)d_in[11];  // [U,1]
  const float* bv     = (const float*)d_in[12];  // [1]
  float* out = (float*)d_out;

  float* ws  = (float*)d_ws;
  float* xz  = ws;                               // B*T*4U   = 4,194,304 f
  float* dec = ws + 4194304;                     // B*T*U    = 1,048,576 f
  float* d1  = ws + 5242880;                     // B*T*U    = 1,048,576 f
  float* e2  = ws + 6291456;                     // B*S*U    = 1,048,576 f

  // 1) one-hot projection gather
  xz_gather_kernel<<<BB * TT, 256, 0, stream>>>(labels, Wx, bvec, xz);

  // 2) e2 = enc @ W2 + b2 (independent of LSTM, launch early)
  gemm_bias_kernel<<<dim3((BB * SS) / 16, 4), 256, 0, stream>>>(enc, W2, b2, e2);

  // 3) sequential LSTM (2 blocks x 16 batch rows, WMMA f32 recurrence)
  lstm_kernel<<<2, 512, 0, stream>>>(xz, Wh, h0, c0, dec);

  // 4) d1 = dec @ W1 + b1
  gemm_bias_kernel<<<dim3((BB * TT) / 16, 4), 256, 0, stream>>>(dec, W1, b1, d1);

  // 5) fused score + softmax, transposed store [B,S,T]
  attn_softmax_kernel<<<BB * TT, 256, 0, stream>>>(d1, e2, V, bv, out);
}